// Cost_Function_39170101739566
// MI455X (gfx1250) — compile-verified
//
#include <hip/hip_runtime.h>
#include <stdint.h>
#include <stddef.h>

#define BEV 200
#define BB  32
#define NN  4096
#define TT  8
#define SPLIT 2
#define K2_THREADS 512
#define NPT (NN / (SPLIT * K2_THREADS))   // 4 trajectories per thread in K2
#define TILE_F4 ((BEV*BEV)/4)             // 10000 float4 chunks per map
#define F4_ITERS ((TILE_F4 + K2_THREADS - 1) / K2_THREADS)  // 20

__device__ __forceinline__ int clampi(int v, int hi) {
    v = v < 0 ? 0 : v;
    return v > hi ? hi : v;
}

// ---------------- K1: sem = semantic_pred * mask(softmax(drivable)[:,1]) ----------------
__global__ void k_sem_mask(const float* __restrict__ spred,
                           const float* __restrict__ drv,
                           float* __restrict__ sem) {
    int i = blockIdx.x * blockDim.x + threadIdx.x;
    const int total = BB * TT * BEV * BEV;
    if (i >= total) return;
    int c = i % BEV;
    int r = (i / BEV) % BEV;
    int b = i / (TT * BEV * BEV);
    int pix = ((b * 2) * BEV + r) * BEV + c;
    float x0 = drv[pix];
    float x1 = drv[pix + BEV * BEV];
    // softmax over 2 channels, channel 1 == sigmoid(x1 - x0)
    float sig = 1.0f / (1.0f + expf(x0 - x1));
    float m = (sig < 0.5f) ? 0.0f : sig;
    sem[i] = spred[i] * m;
}

// ---------------- K2: headway partials via async LDS staging ----------------
// rc is an 8x4 row-major meshgrid: rc[k] = (rowval[k/4], colval[k%4]).
__global__ void __launch_bounds__(K2_THREADS)
k_headway(const float* __restrict__ sem,
          const float* __restrict__ trajs,
          const int* __restrict__ rc,
          float* __restrict__ hw) {
    __shared__ float tile[BEV * BEV];   // 160000 bytes
    __shared__ float rcRowV[8];
    __shared__ float rcColV[4];

    const int tid   = threadIdx.x;
    const int chunk = blockIdx.x;   // n-chunk
    const int t     = blockIdx.y;
    const int b     = blockIdx.z;

    if (tid < 8) rcRowV[tid] = (float)rc[2 * (4 * tid)];      // rows: rc[4i][0]
    if (tid < 4) rcColV[tid] = (float)rc[2 * tid + 1];        // cols: rc[j][1]

    // ---- async stage sem[b,t] (200x200 f32) into LDS: CDNA5 ASYNCcnt path ----
    const float* src = sem + ((size_t)(b * TT + t)) * (BEV * BEV);
    uint32_t tbase = (uint32_t)(uintptr_t)(&tile[0]);
#pragma unroll
    for (int j = 0; j < F4_ITERS; ++j) {
        int i = tid + j * K2_THREADS;
        if (i < TILE_F4) {
            uint32_t loff = tbase + (uint32_t)i * 16u;
            const float* g = src + (size_t)i * 4;
            asm volatile("global_load_async_to_lds_b128 %0, %1, off"
                         :: "v"(loff), "v"(g) : "memory");
        }
    }
    asm volatile("s_wait_asynccnt 0x0" ::: "memory");
    __syncthreads();

    float* hwp = hw + ((size_t)(b * TT + t)) * NN;

#pragma unroll
    for (int j = 0; j < NPT; ++j) {
        int n = chunk * (NN / SPLIT) + j * K2_THREADS + tid;
        const float* p = trajs + (((size_t)(b * NN + n)) * TT + t) * 2;
        float x = p[0];
        float y = p[1];
        // pts = (traj + [0, HEADWAY_L]) / DX + rc ; DX = 0.5 so /DX == *2 (exact)
        float fx = x * 2.0f;
        float fy = (y + 10.0f) * 2.0f;

        int roff[8];
#pragma unroll
        for (int i = 0; i < 8; ++i)
            roff[i] = clampi(__float2int_rz(fy + rcRowV[i]), BEV - 1) * BEV;
        int cidx[4];
#pragma unroll
        for (int i = 0; i < 4; ++i)
            cidx[i] = clampi(__float2int_rz(fx + rcColV[i]), BEV - 1);

        float s = 0.0f;
#pragma unroll
        for (int i = 0; i < 8; ++i)
#pragma unroll
            for (int k = 0; k < 4; ++k)
                s += tile[roff[i] + cidx[k]];
        hwp[n] = s;
    }
}

// ---------------- comfort (shared by K3 and fallback) ----------------
__device__ __forceinline__ float comfort_term(const float* xs, const float* ys) {
    float dxv[TT], dyv[TT], sp[TT];
    float px = 0.0f, py = 0.0f;
#pragma unroll
    for (int t = 0; t < TT; ++t) {
        dxv[t] = xs[t] - px;
        dyv[t] = ys[t] - py;
        px = xs[t]; py = ys[t];
        sp[t] = sqrtf(dxv[t] * dxv[t] + dyv[t] * dyv[t]) * 2.0f;  // speed = |disp|/DT
    }
    float mLat = 0.0f, mLon = 0.0f;
    float ea[TT - 1];
#pragma unroll
    for (int t = 0; t < TT - 1; ++t) {
        float ax = (dxv[t + 1] - dxv[t]) * 4.0f;   // == diff(disp/DT)/DT bitwise
        float ay = (dyv[t + 1] - dyv[t]) * 4.0f;
        mLat = fmaxf(mLat, fabsf(ax));
        mLon = fmaxf(mLon, fabsf(ay));
        ea[t] = (sp[t + 1] - sp[t]) * 2.0f;
    }
    float mJ = 0.0f;
#pragma unroll
    for (int t = 0; t < TT - 2; ++t)
        mJ = fmaxf(mJ, fabsf((ea[t + 1] - ea[t]) * 2.0f));
    float lat = fminf(fmaxf(mLat - 3.0f, 0.0f), 30.0f);
    float lon = fminf(fmaxf(mLon - 3.0f, 0.0f), 30.0f);
    float jk  = fminf(fmaxf(mJ  - 1.0f, 0.0f), 20.0f);
    return lat * lat + lon * lon + jk * jk;
}

// ---------------- K3: comfort + cost-volume gathers + headway reduction ----------------
__global__ void k_final(const float* __restrict__ trajs,
                        const float* __restrict__ cost,
                        const float* __restrict__ hw,
                        float* __restrict__ out) {
    int idx = blockIdx.x * blockDim.x + threadIdx.x;
    if (idx >= BB * NN) return;
    int b = idx / NN;
    int n = idx % NN;

    const float* p = trajs + (size_t)idx * TT * 2;
    float xs[TT], ys[TT];
#pragma unroll
    for (int t = 0; t < TT; ++t) { xs[t] = p[2 * t]; ys[t] = p[2 * t + 1]; }

    float acc = comfort_term(xs, ys);

#pragma unroll
    for (int t = 0; t < TT; ++t) {
        // (traj - BX)/DX with BX=-49.75, DX=0.5
        int yi = clampi(__float2int_rz((ys[t] + 49.75f) * 2.0f), BEV - 1);
        int xi = clampi(__float2int_rz((xs[t] + 49.75f) * 2.0f), BEV - 1);
        float cv = cost[(((size_t)b * TT + t) * BEV + yi) * BEV + xi];
        cv = fminf(fmaxf(cv, 0.0f), 1000.0f);
        acc += cv;
        acc += hw[((size_t)b * TT + t) * NN + n];
    }
    out[idx] = acc;
}

// ---------------- fallback: fully direct (no workspace) ----------------
__global__ void k_direct(const float* __restrict__ trajs,
                         const float* __restrict__ cost,
                         const float* __restrict__ spred,
                         const float* __restrict__ drv,
                         const int* __restrict__ rc,
                         float* __restrict__ out) {
    int idx = blockIdx.x * blockDim.x + threadIdx.x;
    if (idx >= BB * NN) return;
    int b = idx / NN;

    const float* p = trajs + (size_t)idx * TT * 2;
    float xs[TT], ys[TT];
#pragma unroll
    for (int t = 0; t < TT; ++t) { xs[t] = p[2 * t]; ys[t] = p[2 * t + 1]; }

    float rcR[32], rcC[32];
#pragma unroll
    for (int k = 0; k < 32; ++k) { rcR[k] = (float)rc[2 * k]; rcC[k] = (float)rc[2 * k + 1]; }

    float acc = comfort_term(xs, ys);

    for (int t = 0; t < TT; ++t) {
        int yi = clampi(__float2int_rz((ys[t] + 49.75f) * 2.0f), BEV - 1);
        int xi = clampi(__float2int_rz((xs[t] + 49.75f) * 2.0f), BEV - 1);
        float cv = cost[(((size_t)b * TT + t) * BEV + yi) * BEV + xi];
        acc += fminf(fmaxf(cv, 0.0f), 1000.0f);

        float fx = xs[t] * 2.0f;
        float fy = (ys[t] + 10.0f) * 2.0f;
        for (int k = 0; k < 32; ++k) {
            int r = clampi(__float2int_rz(fy + rcR[k]), BEV - 1);
            int c = clampi(__float2int_rz(fx + rcC[k]), BEV - 1);
            float sv = spred[(((size_t)b * TT + t) * BEV + r) * BEV + c];
            int pix = ((b * 2) * BEV + r) * BEV + c;
            float x0 = drv[pix];
            float x1 = drv[pix + BEV * BEV];
            float sig = 1.0f / (1.0f + expf(x0 - x1));
            acc += sv * ((sig < 0.5f) ? 0.0f : sig);
        }
    }
    out[idx] = acc;
}

extern "C" void kernel_launch(void* const* d_in, const int* in_sizes, int n_in,
                              void* d_out, int out_size, void* d_ws, size_t ws_size,
                              hipStream_t stream) {
    const float* trajs = (const float*)d_in[0];
    const float* cost  = (const float*)d_in[1];
    const float* spred = (const float*)d_in[2];
    const float* drv   = (const float*)d_in[3];
    const int*   rc    = (const int*)d_in[4];
    float* out = (float*)d_out;

    const size_t semElems = (size_t)BB * TT * BEV * BEV;  // 10,240,000
    const size_t hwElems  = (size_t)BB * TT * NN;         //  1,048,576
    const size_t need = (semElems + hwElems) * sizeof(float);

    if (ws_size >= need) {
        float* sem = (float*)d_ws;
        float* hw  = sem + semElems;

        int n1 = (int)((semElems + 255) / 256);
        k_sem_mask<<<n1, 256, 0, stream>>>(spred, drv, sem);

        dim3 g2(SPLIT, TT, BB);
        k_headway<<<g2, K2_THREADS, 0, stream>>>(sem, trajs, rc, hw);

        k_final<<<(BB * NN + 255) / 256, 256, 0, stream>>>(trajs, cost, hw, out);
    } else {
        k_direct<<<(BB * NN + 255) / 256, 256, 0, stream>>>(trajs, cost, spred, drv, rc, out);
    }
}